// LigandGNN_12910671692590
// MI455X (gfx1250) — compile-verified
//
#include <hip/hip_runtime.h>

// ---------------- problem constants (match reference) ----------------
constexpr int NN    = 50000;   // nodes
constexpr int EE    = 800000;  // edges
constexpr int BG    = 512;     // graphs
constexpr int H     = 96;      // hidden
constexpr int DIN   = 5;
constexpr int DOUT  = 192;
constexpr float EPSV = 1e-5f;

typedef __attribute__((ext_vector_type(2))) float v2f;
typedef __attribute__((ext_vector_type(8))) float v8f;

// =====================================================================
// Generic fp32 WMMA GEMM: C[M x Ncol] = A[M x K] * B[K x Ncol]
// One wave32 computes one 16x16 tile; K consumed 4 at a time with
// V_WMMA_F32_16X16X4_F32 (exact fp32, §7.12).  M, Ncol multiples of 16,
// K multiple of 4.  Early exit is wave-uniform -> EXEC all-1s at WMMA.
// =====================================================================
__global__ void k_gemm_wmma_f32(const float* __restrict__ A,
                                const float* __restrict__ Bm,
                                float* __restrict__ C,
                                int M, int K, int Ncol) {
    const int wavesPerBlock = blockDim.x >> 5;
    const int wave   = blockIdx.x * wavesPerBlock + (threadIdx.x >> 5);
    const int ctiles = Ncol >> 4;
    const int rt = wave / ctiles;
    const int ct = wave % ctiles;
    if (rt * 16 >= M) return;                 // wave-uniform

    const int lane = threadIdx.x & 31;
    const int half = lane >> 4;               // 0: K 0/1 ; 1: K 2/3
    const int lid  = lane & 15;
    const int arow = rt * 16 + lid;           // A row for this lane
    const int bcol = ct * 16 + lid;           // B/C col for this lane

    v8f acc = {};
    for (int k0 = 0; k0 < K; k0 += 4) {
        const int ka = k0 + half * 2;
        v2f a, b;
        a.x = A[arow * K + ka];
        a.y = A[arow * K + ka + 1];
        b.x = Bm[ka * Ncol + bcol];
        b.y = Bm[(ka + 1) * Ncol + bcol];
        acc = __builtin_amdgcn_wmma_f32_16x16x4_f32(
            /*neg_a=*/false, a, /*neg_b=*/false, b,
            /*c_mod=*/(short)0, acc, /*reuse_a=*/false, /*reuse_b=*/false);
    }
    // C/D layout: vgpr i -> row (rt*16 + half*8 + i), col bcol
    const int rbase = rt * 16 + half * 8;
#pragma unroll
    for (int i = 0; i < 8; ++i)
        C[(rbase + i) * Ncol + bcol] = acc[i];
}

// ---------------- input layer: h = relu(x @ W_in + b_in) -------------
__global__ void k_input(const float* __restrict__ x,
                        const float* __restrict__ Win,
                        const float* __restrict__ bin,
                        float* __restrict__ h) {
    int idx = blockIdx.x * blockDim.x + threadIdx.x;
    if (idx >= NN * H) return;
    int n = idx / H, c = idx % H;
    float acc = bin[c];
#pragma unroll
    for (int k = 0; k < DIN; ++k)
        acc = fmaf(x[n * DIN + k], Win[k * H + c], acc);
    h[idx] = fmaxf(acc, 0.0f);
}

// ---------------- degree / dinv --------------------------------------
__global__ void k_deg(const long long* __restrict__ dst, float* __restrict__ deg) {
    int e = blockIdx.x * blockDim.x + threadIdx.x;
    if (e >= EE) return;
    atomicAdd(&deg[(int)dst[e]], 1.0f);
}
__global__ void k_dinv(float* __restrict__ deg) {
    int n = blockIdx.x * blockDim.x + threadIdx.x;
    if (n >= NN) return;
    deg[n] = rsqrtf(deg[n] + 1.0f);          // self-loop included
}

// ---------------- edge scatter: agg[dst] += hw[src]*coef -------------
// One thread per (edge, 4-channel group): float4 gather + 4 atomics.
// agg (19.2MB) is L2-resident on MI455X (192MB L2).
__global__ void k_agg(const long long* __restrict__ src,
                      const long long* __restrict__ dst,
                      const float* __restrict__ dinv,
                      const float* __restrict__ hw,
                      float* __restrict__ agg) {
    const int groups = H / 4;                 // 24
    long long idx = (long long)blockIdx.x * blockDim.x + threadIdx.x;
    if (idx >= (long long)EE * groups) return;
    int e = (int)(idx / groups);
    int q = (int)(idx % groups) * 4;
    int s = (int)src[e], d = (int)dst[e];
    float coef = dinv[s] * dinv[d];
    const float4 v = *(const float4*)(hw + (size_t)s * H + q);
    float* ap = agg + (size_t)d * H + q;
    atomicAdd(ap + 0, v.x * coef);
    atomicAdd(ap + 1, v.y * coef);
    atomicAdd(ap + 2, v.z * coef);
    atomicAdd(ap + 3, v.w * coef);
}

// ---------------- self-loop + bias: agg += hw*dinv^2 + b -------------
__global__ void k_combine(float* __restrict__ agg,
                          const float* __restrict__ hw,
                          const float* __restrict__ dinv,
                          const float* __restrict__ bvec) {
    int idx = blockIdx.x * blockDim.x + threadIdx.x;
    if (idx >= NN * H) return;
    int n = idx / H, c = idx % H;
    float di = dinv[n];
    agg[idx] = fmaf(hw[idx], di * di, agg[idx]) + bvec[c];
}

// ---------------- batch-norm stats (per-channel sum/sumsq) -----------
// blockDim = 96 (one thread per channel), each block reduces ROWS rows
// locally, then one atomicAdd per (block, channel).
constexpr int BN_ROWS = 128;
__global__ void k_bnstats(const float* __restrict__ v, float* __restrict__ stats) {
    int c  = threadIdx.x;                     // 0..95
    int r0 = blockIdx.x * BN_ROWS;
    int r1 = min(r0 + BN_ROWS, NN);
    float s = 0.0f, s2 = 0.0f;
    for (int r = r0; r < r1; ++r) {
        float t = v[(size_t)r * H + c];
        s += t; s2 += t * t;
    }
    atomicAdd(&stats[c], s);
    atomicAdd(&stats[H + c], s2);
}

// ---------------- BN apply + relu + residual (into h) ----------------
__global__ void k_bnapply(float* __restrict__ h,
                          const float* __restrict__ agg,
                          const float* __restrict__ stats,
                          const float* __restrict__ g,
                          const float* __restrict__ be) {
    int idx = blockIdx.x * blockDim.x + threadIdx.x;
    if (idx >= NN * H) return;
    int c = idx % H;
    float inv_n = 1.0f / (float)NN;
    float mean  = stats[c] * inv_n;
    float var   = stats[H + c] * inv_n - mean * mean;
    float t = g[c] * (agg[idx] - mean) * rsqrtf(var + EPSV) + be[c];
    h[idx] = fmaxf(t, 0.0f) + h[idx];         // relu + residual
}

// ---------------- pooling --------------------------------------------
__global__ void k_count(const long long* __restrict__ batch, float* __restrict__ counts) {
    int n = blockIdx.x * blockDim.x + threadIdx.x;
    if (n >= NN) return;
    atomicAdd(&counts[(int)batch[n]], 1.0f);
}
__global__ void k_pool(const long long* __restrict__ batch,
                       const float* __restrict__ h, float* __restrict__ pooled) {
    int idx = blockIdx.x * blockDim.x + threadIdx.x;
    if (idx >= NN * H) return;
    int n = idx / H, c = idx % H;
    atomicAdd(&pooled[(size_t)batch[n] * H + c], h[idx]);
}
__global__ void k_pooldiv(float* __restrict__ pooled, const float* __restrict__ counts) {
    int idx = blockIdx.x * blockDim.x + threadIdx.x;
    if (idx >= BG * H) return;
    pooled[idx] /= fmaxf(counts[idx / H], 1.0f);
}

// ---------------- final layernorm over D_OUT per row -----------------
__global__ void k_ln(const float* __restrict__ obuf,
                     const float* __restrict__ b_out,
                     const float* __restrict__ ln_g,
                     const float* __restrict__ ln_b,
                     float* __restrict__ dout) {
    int b = blockIdx.x * blockDim.x + threadIdx.x;
    if (b >= BG) return;
    float s = 0.0f, s2 = 0.0f;
    for (int c = 0; c < DOUT; ++c) {
        float v = obuf[b * DOUT + c] + b_out[c];
        s += v; s2 += v * v;
    }
    float mu  = s / (float)DOUT;
    float var = s2 / (float)DOUT - mu * mu;
    float rs  = rsqrtf(var + EPSV);
    for (int c = 0; c < DOUT; ++c) {
        float v = obuf[b * DOUT + c] + b_out[c];
        dout[b * DOUT + c] = ln_g[c] * (v - mu) * rs + ln_b[c];
    }
}

// =====================================================================
static inline int cdiv(long long a, long long b) { return (int)((a + b - 1) / b); }

extern "C" void kernel_launch(void* const* d_in, const int* in_sizes, int n_in,
                              void* d_out, int out_size, void* d_ws, size_t ws_size,
                              hipStream_t stream) {
    // inputs in setup_inputs() order
    const float*     x      = (const float*)d_in[0];
    const long long* eidx   = (const long long*)d_in[1];   // [2, E] int64
    const long long* batch  = (const long long*)d_in[2];   // [N] int64
    const float*     W_in   = (const float*)d_in[3];
    const float*     b_in   = (const float*)d_in[4];
    const float*     W1     = (const float*)d_in[5];
    const float*     b1     = (const float*)d_in[6];
    const float*     g1     = (const float*)d_in[7];
    const float*     be1    = (const float*)d_in[8];
    const float*     W2     = (const float*)d_in[9];
    const float*     b2     = (const float*)d_in[10];
    const float*     g2     = (const float*)d_in[11];
    const float*     be2    = (const float*)d_in[12];
    const float*     W_out  = (const float*)d_in[13];
    const float*     b_out  = (const float*)d_in[14];
    const float*     ln_g   = (const float*)d_in[15];
    const float*     ln_b   = (const float*)d_in[16];
    float* out = (float*)d_out;

    const long long* srcv = eidx;            // row 0
    const long long* dstv = eidx + EE;       // row 1

    // -------- workspace layout (256B aligned chunks) --------
    char* ws = (char*)d_ws;
    size_t off = 0;
    auto alloc = [&](size_t bytes) {
        char* p = ws + off;
        off += (bytes + 255) & ~(size_t)255;
        return p;
    };
    float* dinv   = (float*)alloc((size_t)NN * sizeof(float));
    float* h      = (float*)alloc((size_t)NN * H * sizeof(float));
    float* hw     = (float*)alloc((size_t)NN * H * sizeof(float));
    float* agg    = (float*)alloc((size_t)NN * H * sizeof(float));
    float* stats  = (float*)alloc(2 * H * sizeof(float));
    float* counts = (float*)alloc(BG * sizeof(float));
    float* pooled = (float*)alloc((size_t)BG * H * sizeof(float));
    float* obuf   = (float*)alloc((size_t)BG * DOUT * sizeof(float));
    (void)ws_size;

    const int TB = 256;
    const int NH = NN * H;

    // ---- input projection ----
    k_input<<<cdiv(NH, TB), TB, 0, stream>>>(x, W_in, b_in, h);

    // ---- degrees (shared by both layers) ----
    hipMemsetAsync(dinv, 0, (size_t)NN * sizeof(float), stream);
    k_deg<<<cdiv(EE, TB), TB, 0, stream>>>(dstv, dinv);
    k_dinv<<<cdiv(NN, TB), TB, 0, stream>>>(dinv);

    // ---- two GCN layers ----
    const float* Ws[2]  = {W1, W2};
    const float* bs[2]  = {b1, b2};
    const float* gs[2]  = {g1, g2};
    const float* bes[2] = {be1, be2};
    const int gemm_waves_h = (NN / 16) * (H / 16);           // 3125*6
    for (int l = 0; l < 2; ++l) {
        // hw = h @ W   (fp32 WMMA)
        k_gemm_wmma_f32<<<cdiv((long long)gemm_waves_h * 32, TB), TB, 0, stream>>>(
            h, Ws[l], hw, NN, H, H);
        // agg = scatter(hw[src]*coef -> dst)
        hipMemsetAsync(agg, 0, (size_t)NH * sizeof(float), stream);
        k_agg<<<cdiv((long long)EE * (H / 4), TB), TB, 0, stream>>>(
            srcv, dstv, dinv, hw, agg);
        // agg += hw*dinv^2 + b
        k_combine<<<cdiv(NH, TB), TB, 0, stream>>>(agg, hw, dinv, bs[l]);
        // batch-norm stats
        hipMemsetAsync(stats, 0, 2 * H * sizeof(float), stream);
        k_bnstats<<<cdiv(NN, BN_ROWS), H, 0, stream>>>(agg, stats);
        // h = relu(bn(agg)) + h
        k_bnapply<<<cdiv(NH, TB), TB, 0, stream>>>(h, agg, stats, gs[l], bes[l]);
    }

    // ---- global mean pool ----
    hipMemsetAsync(counts, 0, BG * sizeof(float), stream);
    hipMemsetAsync(pooled, 0, (size_t)BG * H * sizeof(float), stream);
    k_count<<<cdiv(NN, TB), TB, 0, stream>>>(batch, counts);
    k_pool<<<cdiv(NH, TB), TB, 0, stream>>>(batch, h, pooled);
    k_pooldiv<<<cdiv(BG * H, TB), TB, 0, stream>>>(pooled, counts);

    // ---- output projection (fp32 WMMA) + layernorm ----
    const int gemm_waves_o = (BG / 16) * (DOUT / 16);        // 32*12
    k_gemm_wmma_f32<<<cdiv((long long)gemm_waves_o * 32, TB), TB, 0, stream>>>(
        pooled, W_out, obuf, BG, H, DOUT);
    k_ln<<<cdiv(BG, TB), TB, 0, stream>>>(obuf, b_out, ln_g, ln_b, out);
    (void)out_size; (void)n_in; (void)in_sizes;
}